// mLSTMCell_78537771975160
// MI455X (gfx1250) — compile-verified
//
#include <hip/hip_runtime.h>
#include <math.h>

#define SQ 2048
#define NHEAD 8
#define DHD 128

// output tuple layout (floats): c | n | m | output
#define OUT_C 0
#define OUT_N 131072            // 8*128*128
#define OUT_M 132096            // + 8*128
#define OUT_O 132104            // + 8

typedef unsigned int u32;
typedef unsigned long long u64;
typedef __attribute__((ext_vector_type(2))) float v2f;
typedef __attribute__((ext_vector_type(8))) float v8f;
typedef __attribute__((ext_vector_type(4))) u32 v4u;
typedef __attribute__((ext_vector_type(8))) int v8i;
typedef __attribute__((ext_vector_type(4))) int v4i;

__device__ __forceinline__ v8f wmma4(v2f a, v2f b, v8f c) {
  // D = A(16x4) * B(4x16) + C, full fp32.
  return __builtin_amdgcn_wmma_f32_16x16x4_f32(false, a, false, b, (short)0, c,
                                               false, false);
}

__device__ __forceinline__ v8f zero8() {
  v8f z = {0.f, 0.f, 0.f, 0.f, 0.f, 0.f, 0.f, 0.f};
  return z;
}

// ---------------------------------------------------------------------------
// TDM: DMA a 16x128 f32 tile (row stride 128 floats in memory) into LDS,
// inserting 4 dwords of padding after every 128 dwords -> LDS row stride 132.
// D# per CDNA5 ISA.
// ---------------------------------------------------------------------------
__device__ __forceinline__ void tdm_load_16x128(const float* gsrc,
                                                u32 lds_addr) {
  u64 ga = (u64)(uintptr_t)gsrc;
  v4u g0;
  g0.x = 1u;                                           // count=1 (valid D#)
  g0.y = lds_addr;                                     // lds_addr
  g0.z = (u32)ga;                                      // global_addr[31:0]
  g0.w = ((u32)(ga >> 32) & 0x01ffffffu) | 0x80000000u;  // addr[56:32]|type=2
  v8i g1;
  g1[0] = (int)((2u << 16)    // data_size = 4 bytes
                | (1u << 20)  // pad_enable
                | (6u << 22)  // pad_interval = 128 dwords
                | (3u << 25));  // pad_amount = 4 dwords
  g1[1] = (int)(128u << 16);  // tensor_dim0[15:0]=128 (atomic barrier addr=0)
  g1[2] = (int)(16u << 16);   // tensor_dim0 hi=0 | tensor_dim1[15:0]=16
  g1[3] = (int)(128u << 16);  // tensor_dim1 hi=0 | tile_dim0=128
  g1[4] = 16;                 // tile_dim1=16, tile_dim2=0
  g1[5] = 128;                // tensor_dim0_stride = 128 elements
  g1[6] = 0;
  g1[7] = 0;
  v4i z4 = {0, 0, 0, 0};
#if defined(__clang_major__) && (__clang_major__ >= 23)
  v8i z8 = {0, 0, 0, 0, 0, 0, 0, 0};
  __builtin_amdgcn_tensor_load_to_lds(g0, g1, z4, z4, z8, 0);
#else
  __builtin_amdgcn_tensor_load_to_lds(g0, g1, z4, z4, 0);
#endif
}

// ---------------------------------------------------------------------------
// Kernel 1: gate projections  i = qkv@wi+bi , log_f = -softplus(-(qkv@wf+bf))
// flat index t*8+c  ==  head-major [h][s] (since t*8+c == h*2048+s)
// ---------------------------------------------------------------------------
__global__ __launch_bounds__(256) void k_gates(
    const float* __restrict__ q, const float* __restrict__ k,
    const float* __restrict__ v, const float* __restrict__ wik,
    const float* __restrict__ wib, const float* __restrict__ wfk,
    const float* __restrict__ wfb, float* __restrict__ ws_i,
    float* __restrict__ ws_lf) {
  const int idx = blockIdx.x * 256 + threadIdx.x;  // 0..16383
  const int t = idx >> 3, c = idx & 7;
  const float* qr = q + t * 1024;
  const float* kr = k + t * 1024;
  const float* vr = v + t * 1024;
  float ai = wib[c], af = wfb[c];
  for (int kk = 0; kk < 1024; ++kk) {
    float x = qr[kk];
    ai += x * wik[kk * 8 + c];
    af += x * wfk[kk * 8 + c];
  }
  for (int kk = 0; kk < 1024; ++kk) {
    float x = kr[kk];
    ai += x * wik[(1024 + kk) * 8 + c];
    af += x * wfk[(1024 + kk) * 8 + c];
  }
  for (int kk = 0; kk < 1024; ++kk) {
    float x = vr[kk];
    ai += x * wik[(2048 + kk) * 8 + c];
    af += x * wfk[(2048 + kk) * 8 + c];
  }
  ws_i[idx] = ai;
  // -softplus(-f) = min(f,0) - log1p(exp(-|f|))
  ws_lf[idx] = fminf(af, 0.f) - log1pf(expf(-fabsf(af)));
}

// ---------------------------------------------------------------------------
// Kernel 2: per-head scans: lfc = cumsum(log_f), g = i - lfc,
//   mx = max(segmented running max of g, prev_m), ep = cumsum(mask),
//   plus final-state scalars mM, prev_decay, m.
// ---------------------------------------------------------------------------
__global__ __launch_bounds__(256) void k_scan(
    const int* __restrict__ mask, const float* __restrict__ prev_m,
    const float* __restrict__ ws_i, const float* __restrict__ ws_lf,
    float* __restrict__ ws_lfc, float* __restrict__ ws_g,
    float* __restrict__ ws_mx, int* __restrict__ ws_ep,
    float* __restrict__ ws_mM, float* __restrict__ ws_pd,
    float* __restrict__ out_m) {
  __shared__ float sF[256];
  __shared__ int sI[256];
  __shared__ int sCE[256];
  __shared__ float sCV[256];
  __shared__ float sRed[256];
  __shared__ float sFinal;

  const int h = blockIdx.x;
  const int tid = threadIdx.x;
  const int base = tid * 8;
  const float pm = prev_m[h];

  // ---- ep = inclusive cumsum(mask)
  int epl[8];
  {
    int run = 0;
#pragma unroll
    for (int i = 0; i < 8; ++i) {
      run += mask[base + i];
      epl[i] = run;
    }
    sI[tid] = run;
  }
  __syncthreads();
  if (tid == 0) {
    int r = 0;
    for (int j = 0; j < 256; ++j) {
      int t = sI[j];
      sI[j] = r;
      r += t;
    }
  }
  __syncthreads();
  {
    int o = sI[tid];
#pragma unroll
    for (int i = 0; i < 8; ++i) epl[i] += o;
  }
  if (h == 0) {
#pragma unroll
    for (int i = 0; i < 8; ++i) ws_ep[base + i] = epl[i];
  }
  __syncthreads();

  // ---- lfc scan
  float lfl[8];
  {
    float run = 0.f;
#pragma unroll
    for (int i = 0; i < 8; ++i) {
      run += ws_lf[h * SQ + base + i];
      lfl[i] = run;
    }
    sF[tid] = run;
  }
  __syncthreads();
  if (tid == 0) {
    float r = 0.f;
    for (int j = 0; j < 256; ++j) {
      float t = sF[j];
      sF[j] = r;
      r += t;
    }
  }
  __syncthreads();
  float gl[8];
  float gmaxl = -3.0e38f;
  {
    float o = sF[tid];
#pragma unroll
    for (int i = 0; i < 8; ++i) {
      lfl[i] += o;
      ws_lfc[h * SQ + base + i] = lfl[i];
      float gg = ws_i[h * SQ + base + i] - lfl[i];
      gl[i] = gg;
      ws_g[h * SQ + base + i] = gg;
      gmaxl = fmaxf(gmaxl, gg);
    }
  }
  if (tid == 255) sFinal = lfl[7];

  // ---- segmented running max of g (segment = equal ep run; ep monotonic)
  float sml[8];
  {
    int ce = epl[0];
    float cv = gl[0];
    sml[0] = cv;
#pragma unroll
    for (int i = 1; i < 8; ++i) {
      if (epl[i] == ce)
        cv = fmaxf(cv, gl[i]);
      else {
        ce = epl[i];
        cv = gl[i];
      }
      sml[i] = cv;
    }
    sCE[tid] = ce;
    sCV[tid] = cv;
  }
  sRed[tid] = gmaxl;
  __syncthreads();
  if (tid == 0) {
    int ce = -1;
    float cv = -3.0e38f;
    for (int j = 0; j < 256; ++j) {
      int te = sCE[j];
      float tv = sCV[j];
      sCE[j] = ce;  // incoming carry for chunk j
      sCV[j] = cv;
      if (te == ce) tv = fmaxf(tv, cv);  // chunk trailing run connects
      ce = te;
      cv = tv;
    }
  }
  __syncthreads();
  {
    int cE = sCE[tid];
    float cV = sCV[tid];
#pragma unroll
    for (int i = 0; i < 8; ++i) {
      float val = sml[i];
      if (epl[i] == cE) val = fmaxf(val, cV);
      ws_mx[h * SQ + base + i] = fmaxf(val, pm);
    }
  }

  if (tid == 0) {
    float gm = -3.0e38f;
    for (int j = 0; j < 256; ++j) gm = fmaxf(gm, sRed[j]);
    float mM = fmaxf(pm, gm);
    ws_mM[h] = mM;
    ws_pd[h] = expf(pm - mM);  // prev_decay
    out_m[h] = sFinal + mM;    // m
  }
}

// ---------------------------------------------------------------------------
// Kernel 3: main pass. grid = 8 heads * 32 row-blocks(64), 4 waves * 16 rows.
// TDM double-buffered K/V (and prev_c) tiles -> LDS; f32 WMMA 16x16x4.
// ---------------------------------------------------------------------------
__global__ __launch_bounds__(128) void k_attn(
    const float* __restrict__ q, const float* __restrict__ k,
    const float* __restrict__ v, const float* __restrict__ prev_c,
    const float* __restrict__ prev_n, const float* __restrict__ prev_m,
    const float* __restrict__ ws_lfc, const float* __restrict__ ws_g,
    const float* __restrict__ ws_mx, const int* __restrict__ ws_ep,
    float* __restrict__ out) {
  __shared__ float Ks[2][16 * 132];
  __shared__ float Vs[2][16 * 132];
  __shared__ float Ws[4][16 * 20];

  const int blk = blockIdx.x;
  const int h = blk >> 5;
  const int rb = blk & 31;
  const int s0b = rb << 6;
  const int tid = threadIdx.x;
  const int wave = tid >> 5;
  const int lane = tid & 31;
  const int lo = lane & 15;
  const int hi = lane >> 4;
  const int koff = hi << 1;

  const float* qh = q + h * (SQ * DHD);
  const float* kh = k + h * (SQ * DHD);
  const float* vh = v + h * (SQ * DHD);
  const float* lfc = ws_lfc + h * SQ;
  const float* g = ws_g + h * SQ;
  const float* mxa = ws_mx + h * SQ;
  const float pm = prev_m[h];

  const int s0w = s0b + (wave << 4);
  const int srl = s0w + lo;

  float r_lfc = lfc[srl];
  float r_mx = mxa[srl];
  float r_dec = expf(pm - r_mx);      // decay[s]
  float r_thr = expf(-r_lfc - r_mx);  // exp(-stab[s])
  int r_ep = ws_ep[srl];

  // broadcast per-row scalars into C/D layout order (row = r + 8*hi)
  float mx8[8], dec8[8], thr8[8];
  int ep8[8], sr8[8];
#pragma unroll
  for (int r = 0; r < 8; ++r) {
    int idx = r + (hi << 3);
    mx8[r] = __shfl(r_mx, idx, 32);
    dec8[r] = __shfl(r_dec, idx, 32);
    thr8[r] = __shfl(r_thr, idx, 32);
    ep8[r] = __shfl(r_ep, idx, 32);
    sr8[r] = s0w + idx;
  }

  // Q A-fragments (lane<16: row lo, K pair 4c..4c+1; lane>=16: 4c+2..4c+3)
  const float qscale = 0.088388347648318447f;  // 1/sqrt(128)
  v2f qa[32];
  {
    const float* qr = qh + srl * DHD;
#pragma unroll
    for (int c = 0; c < 32; ++c) {
      v2f t = *(const v2f*)(qr + 4 * c + koff);
      qa[c] = t * qscale;
    }
  }

  v8f acc[8];
#pragma unroll
  for (int nt = 0; nt < 8; ++nt) acc[nt] = zero8();

  // ---- Phase A: acc = Q @ prev_c, prev_c staged by TDM, double buffered
  const float* pc = prev_c + h * (DHD * DHD);
  if (wave == 0) tdm_load_16x128(pc, (u32)(uintptr_t)&Ks[0][0]);
  for (int kk = 0; kk < 8; ++kk) {
    if (wave == 0) {
      if (kk < 7) {
        tdm_load_16x128(pc + (kk + 1) * 16 * DHD,
                        (u32)(uintptr_t)&Ks[(kk + 1) & 1][0]);
        __builtin_amdgcn_s_wait_tensorcnt(1);
      } else {
        __builtin_amdgcn_s_wait_tensorcnt(0);
      }
    }
    __syncthreads();
    const float* T = &Ks[kk & 1][0];
#pragma unroll
    for (int nt = 0; nt < 8; ++nt) {
      v8f a = acc[nt];
#pragma unroll
      for (int c = 0; c < 4; ++c) {
        v2f b;
        b.x = T[(4 * c + koff) * 132 + (nt << 4) + lo];
        b.y = T[(4 * c + koff + 1) * 132 + (nt << 4) + lo];
        a = wmma4(qa[4 * kk + c], b, a);
      }
      acc[nt] = a;
    }
    __syncthreads();
  }
  // inter_c = decay[s] * (Q @ prev_c)
#pragma unroll
  for (int nt = 0; nt < 8; ++nt) {
    v8f a = acc[nt];
#pragma unroll
    for (int r = 0; r < 8; ++r) a[r] *= dec8[r];
    acc[nt] = a;
  }

  // inter_n = decay[s] * (Q @ prev_n)
  float intn8[8];
  {
    const float* pn = prev_n + h * DHD;
    float s = 0.f;
#pragma unroll
    for (int c = 0; c < 32; ++c)
      s += qa[c].x * pn[4 * c + koff] + qa[c].y * pn[4 * c + koff + 1];
    s += __shfl_xor(s, 16, 32);
    s *= r_dec;
#pragma unroll
    for (int r = 0; r < 8; ++r) intn8[r] = __shfl(s, r + (hi << 3), 32);
  }

  float rs[8];
#pragma unroll
  for (int r = 0; r < 8; ++r) rs[r] = 0.f;

  // ---- Phase B: j-tile loop, TDM double-buffered K/V
  const int ntile = (s0b >> 4) + 4;  // tiles 0..(s0b+63)/16
  if (wave == 0) {
    tdm_load_16x128(kh, (u32)(uintptr_t)&Ks[0][0]);
    tdm_load_16x128(vh, (u32)(uintptr_t)&Vs[0][0]);
  }
  float* wsm = &Ws[wave][0];
  for (int t = 0; t < ntile; ++t) {
    const int j0 = t << 4;
    if (wave == 0) {
      if (t + 1 < ntile) {
        tdm_load_16x128(kh + (t + 1) * 16 * DHD,
                        (u32)(uintptr_t)&Ks[(t + 1) & 1][0]);
        tdm_load_16x128(vh + (t + 1) * 16 * DHD,
                        (u32)(uintptr_t)&Vs[(t + 1) & 1][0]);
        __builtin_amdgcn_s_wait_tensorcnt(2);
      } else {
        __builtin_amdgcn_s_wait_tensorcnt(0);
      }
    }
    __syncthreads();
    if (j0 <= s0w + 15) {
      const float* KT = &Ks[t & 1][0];
      const float* VT = &Vs[t & 1][0];

      // E = Q @ K^T (16x16)
      v8f e = zero8();
#pragma unroll 8
      for (int c = 0; c < 32; ++c) {
        v2f b = *(const v2f*)(&KT[lo * 132 + 4 * c + koff]);
        e = wmma4(qa[c], b, e);
      }

      // weight: w = E * exp(g[j]-mx[s]) * [ep match & causal]
      const int jc = j0 + lo;
      const float gj = g[jc];
      const int epj = ws_ep[jc];
      float wf[8];
#pragma unroll
      for (int r = 0; r < 8; ++r) {
        float w = 0.f;
        if (epj == ep8[r] && jc <= sr8[r]) w = e[r] * expf(gj - mx8[r]);
        wf[r] = w;
        rs[r] += w;
      }
      // stage W through LDS: C-layout -> A-layout transpose
#pragma unroll
      for (int r = 0; r < 8; ++r) wsm[(r + (hi << 3)) * 20 + lo] = wf[r];
      asm volatile("s_wait_dscnt 0" ::: "memory");  // wave-local LDS RAW

      // acc += W @ V
#pragma unroll
      for (int nt = 0; nt < 8; ++nt) {
        v8f a = acc[nt];
#pragma unroll
        for (int c = 0; c < 4; ++c) {
          v2f aw = *(const v2f*)(&wsm[lo * 20 + 4 * c + koff]);
          v2f b;
          b.x = VT[(4 * c + koff) * 132 + (nt << 4) + lo];
          b.y = VT[(4 * c + koff + 1) * 132 + (nt << 4) + lo];
          a = wmma4(aw, b, a);
        }
        acc[nt] = a;
      }
    }
    __syncthreads();
  }

  // reduce row sums across the 16 lanes of each half
#pragma unroll
  for (int m = 1; m < 16; m <<= 1) {
#pragma unroll
    for (int r = 0; r < 8; ++r) rs[r] += __shfl_xor(rs[r], m, 32);
  }

  float* oph = out + OUT_O + h * (SQ * DHD);
#pragma unroll
  for (int r = 0; r < 8; ++r) {
    float nrm = fmaxf(fabsf(rs[r]) + intn8[r], thr8[r]);
    float sc = 1.f / (nrm + 1e-6f);
    float* orow = oph + sr8[r] * DHD;
#pragma unroll
    for (int nt = 0; nt < 8; ++nt) orow[(nt << 4) + lo] = acc[nt][r] * sc;
  }
}

// ---------------------------------------------------------------------------
// Kernel 4: final state  c = prev_c*pd + K^T diag(kv_decay) V ,
//                        n = prev_n*pd + K^T kv_decay
// one block per head; 8 waves = 8 d1-tiles; TDM double-buffered K/V tiles.
// ---------------------------------------------------------------------------
__global__ __launch_bounds__(256) void k_state(
    const float* __restrict__ k, const float* __restrict__ v,
    const float* __restrict__ prev_c, const float* __restrict__ prev_n,
    const float* __restrict__ ws_g, const float* __restrict__ ws_mM,
    const float* __restrict__ ws_pd, float* __restrict__ out) {
  __shared__ float Ks[2][16 * 132];
  __shared__ float Vs[2][16 * 132];

  const int h = blockIdx.x;
  const int tid = threadIdx.x;
  const int wave = tid >> 5;
  const int lane = tid & 31;
  const int lo = lane & 15, hi = lane >> 4, koff = hi << 1;
  const int d10 = wave << 4;

  const float* kh = k + h * (SQ * DHD);
  const float* vh = v + h * (SQ * DHD);
  const float* g = ws_g + h * SQ;
  const float mM = ws_mM[h];
  const float pd = ws_pd[h];

  v8f acc[8];
#pragma unroll
  for (int nt = 0; nt < 8; ++nt) acc[nt] = zero8();
  float na = 0.f;

  if (wave == 0) {
    tdm_load_16x128(kh, (u32)(uintptr_t)&Ks[0][0]);
    tdm_load_16x128(vh, (u32)(uintptr_t)&Vs[0][0]);
  }
  for (int t = 0; t < 128; ++t) {
    if (wave == 0) {
      if (t < 127) {
        tdm_load_16x128(kh + (t + 1) * 16 * DHD,
                        (u32)(uintptr_t)&Ks[(t + 1) & 1][0]);
        tdm_load_16x128(vh + (t + 1) * 16 * DHD,
                        (u32)(uintptr_t)&Vs[(t + 1) & 1][0]);
        __builtin_amdgcn_s_wait_tensorcnt(2);
      } else {
        __builtin_amdgcn_s_wait_tensorcnt(0);
      }
    }
    __syncthreads();
    const float* KT = &Ks[t & 1][0];
    const float* VT = &Vs[t & 1][0];
    const int sb = t << 4;
#pragma unroll
    for (int c = 0; c < 4; ++c) {
      const int r0 = 4 * c + koff;
      const int sa = sb + r0;
      float kda = expf(g[sa] - mM);      // kv_decay
      float kdb = expf(g[sa + 1] - mM);
      v2f a;
      a.x = KT[r0 * 132 + d10 + lo] * kda;
      a.y = KT[(r0 + 1) * 132 + d10 + lo] * kdb;
      na += a.x + a.y;
#pragma unroll
      for (int nt = 0; nt < 8; ++nt) {
        v2f b;
        b.x = VT[r0 * 132 + (nt << 4) + lo];
        b.y = VT[(r0 + 1) * 132 + (nt << 4) + lo];
        acc[nt] = wmma4(a, b, acc[nt]);
      }
    }
    __syncthreads();
  }

  na += __shfl_xor(na, 16, 32);
  if (lane < 16)
    out[OUT_N + h * DHD + d10 + lo] = prev_n[h * DHD + d10 + lo] * pd + na;

  const float* pc = prev_c + h * (DHD * DHD);
  float* oc = out + OUT_C + h * (DHD * DHD);
#pragma unroll
  for (int r = 0; r < 8; ++r) {
    int d1 = d10 + r + (hi << 3);
#pragma unroll
    for (int nt = 0; nt < 8; ++nt) {
      int cix = d1 * DHD + (nt << 4) + lo;
      oc[cix] = pc[cix] * pd + acc[nt][r];
    }
  }
}

// ---------------------------------------------------------------------------
extern "C" void kernel_launch(void* const* d_in, const int* in_sizes, int n_in,
                              void* d_out, int out_size, void* d_ws,
                              size_t ws_size, hipStream_t stream) {
  const float* q = (const float*)d_in[0];
  const float* k = (const float*)d_in[1];
  const float* v = (const float*)d_in[2];
  const int* mask = (const int*)d_in[3];
  const float* prev_c = (const float*)d_in[4];
  const float* prev_n = (const float*)d_in[5];
  const float* prev_m = (const float*)d_in[6];
  const float* wik = (const float*)d_in[7];
  const float* wib = (const float*)d_in[8];
  const float* wfk = (const float*)d_in[9];
  const float* wfb = (const float*)d_in[10];
  float* out = (float*)d_out;
  float* ws = (float*)d_ws;

  float* ws_i = ws;                 // [8][2048] i_gate
  float* ws_lf = ws + 16384;        // [8][2048] log_f
  float* ws_lfc = ws + 32768;       // [8][2048] cumsum(log_f)
  float* ws_g = ws + 49152;         // [8][2048] i - lfc
  float* ws_mx = ws + 65536;        // [8][2048] max(segmax g, prev_m)
  int* ws_ep = (int*)(ws + 81920);  // [2048]
  float* ws_mM = ws + 83968;        // [8]
  float* ws_pd = ws + 83976;        // [8]

  k_gates<<<64, 256, 0, stream>>>(q, k, v, wik, wib, wfk, wfb, ws_i, ws_lf);
  k_scan<<<8, 256, 0, stream>>>(mask, prev_m, ws_i, ws_lf, ws_lfc, ws_g, ws_mx,
                                ws_ep, ws_mM, ws_pd, out + OUT_M);
  k_attn<<<256, 128, 0, stream>>>(q, k, v, prev_c, prev_n, prev_m, ws_lfc, ws_g,
                                  ws_mx, ws_ep, out);
  k_state<<<8, 256, 0, stream>>>(k, v, prev_c, prev_n, ws_g, ws_mM, ws_pd, out);
}